// GCNWithNoise_7567732375599
// MI455X (gfx1250) — compile-verified
//
#include <hip/hip_runtime.h>
#include <math.h>

typedef __attribute__((ext_vector_type(2))) float v2f;
typedef __attribute__((ext_vector_type(8))) float v8f;

#define DF 64   // feature dim (in = hid = out = 64)

// ---------------------------------------------------------------------------
// Degree / normalization kernels
// ---------------------------------------------------------------------------
__global__ void k_init_deg(float* __restrict__ deg, int N) {
  int i = blockIdx.x * blockDim.x + threadIdx.x;
  if (i < N) deg[i] = 1.0f;                 // self-loop weight pre-added
}

__global__ void k_accum_deg(const int* __restrict__ ei, const float* __restrict__ w,
                            float* __restrict__ deg, int E) {
  int e = blockIdx.x * blockDim.x + threadIdx.x;
  if (e < E) atomicAdd(&deg[ei[E + e]], w[e]);   // dst = edge_index[1][e]
}

__global__ void k_dinv(float* __restrict__ deg, int N) {
  int i = blockIdx.x * blockDim.x + threadIdx.x;
  if (i < N) {
    float d = deg[i];
    deg[i] = (d > 0.0f) ? rsqrtf(d) : 0.0f;      // in-place deg -> dinv
  }
}

// ---------------------------------------------------------------------------
// H = X @ W  via V_WMMA_F32_16X16X4_F32 (exact f32, matches reference numerics)
// One wave computes a 16-row x 64-col slab: 4 accumulator tiles, K stepped by 4.
// W is staged in LDS with K-pairs interleaved so every B fragment is a single
// aligned ds_load_b64 into an even VGPR pair (no v_mov glue).
// ---------------------------------------------------------------------------
__global__ __launch_bounds__(256) void k_gemm_wmma(const float* __restrict__ X,
                                                   const float* __restrict__ W,
                                                   float* __restrict__ H, int N) {
  // lw[(p*64 + n)*2 + q] = W[2p+q][n]   (p = K-pair 0..31, n = col, q = 0/1)
  __shared__ float lw[DF * DF];
  for (int i = threadIdx.x; i < DF * DF; i += 256) {
    const int k = i >> 6;                        // row of W
    const int n = i & 63;                        // col of W
    lw[(((k >> 1) << 6) + n) * 2 + (k & 1)] = W[i];
  }
  __syncthreads();

  const int wave = threadIdx.x >> 5;
  const int lane = threadIdx.x & 31;
  const int half = lane >> 4;                    // 0: lanes 0-15, 1: lanes 16-31
  const int lid  = lane & 15;
  const int row0 = (blockIdx.x * 8 + wave) * 16;

  // A-row this lane feeds (clamped so EXEC stays all-ones; stores are guarded)
  int arow = row0 + lid;
  if (arow > N - 1) arow = N - 1;
  const float* xr = X + (long)arow * DF;

  v8f c0 = {}, c1 = {}, c2 = {}, c3 = {};

  for (int k0 = 0; k0 < DF; k0 += 4) {
    const int p = (k0 >> 1) + half;              // K-pair handled by this half-wave
    // A 16x4 layout: VGPR0 = K(2p), VGPR1 = K(2p+1) -> contiguous b64 load
    const v2f a = *(const v2f*)(xr + 2 * p);
    // B 4x16 layout: same striping; interleaved LDS gives one b64 per tile
    const v2f* wb = (const v2f*)&lw[((p << 6) + lid) * 2];
    const v2f b0 = wb[0];                        // column tile n0 = 0
    const v2f b1 = wb[16];                       // n0 = 16
    const v2f b2 = wb[32];                       // n0 = 32
    const v2f b3 = wb[48];                       // n0 = 48

    c0 = __builtin_amdgcn_wmma_f32_16x16x4_f32(false, a, false, b0, (short)0, c0, false, false);
    c1 = __builtin_amdgcn_wmma_f32_16x16x4_f32(false, a, false, b1, (short)0, c1, false, false);
    c2 = __builtin_amdgcn_wmma_f32_16x16x4_f32(false, a, false, b2, (short)0, c2, false, false);
    c3 = __builtin_amdgcn_wmma_f32_16x16x4_f32(false, a, false, b3, (short)0, c3, false, false);
  }

  // D layout: VGPR r, lanes 0-15 -> (m=r, n=lid); lanes 16-31 -> (m=r+8, n=lid)
  for (int r = 0; r < 8; ++r) {
    const int m = row0 + r + 8 * half;
    if (m < N) {
      float* hr = H + (long)m * DF + lid;
      hr[0]  = c0[r];
      hr[16] = c1[r];
      hr[32] = c2[r];
      hr[48] = c3[r];
    }
  }
}

// ---------------------------------------------------------------------------
// agg[i][:] = b[:]
// ---------------------------------------------------------------------------
__global__ void k_init_agg(float* __restrict__ agg, const float* __restrict__ b, long tot) {
  long i = (long)blockIdx.x * blockDim.x + threadIdx.x;
  if (i < tot) agg[i] = b[i & (DF - 1)];
}

// ---------------------------------------------------------------------------
// Edge scatter: agg[dst] += h[src] * (dinv[src]*w*dinv[dst]).
// 16 threads per edge, one float4 (16B) gather + 4 f32 atomics each.
// This is the bandwidth-dominant phase (~0.4 GB + 51M atomics per layer).
// ---------------------------------------------------------------------------
__global__ void k_scatter_edges(const int* __restrict__ ei, const float* __restrict__ w,
                                const float* __restrict__ dinv, const float* __restrict__ H,
                                float* __restrict__ agg, int E) {
  long t = (long)blockIdx.x * blockDim.x + threadIdx.x;
  long e = t >> 4;
  if (e >= E) return;
  const int f = (int)(t & 15) * 4;
  const int s = ei[e];
  const int d = ei[E + e];
  const float nrm = dinv[s] * w[e] * dinv[d];
  const float4 hv = *(const float4*)(H + (long)s * DF + f);
  float* ap = agg + (long)d * DF + f;
  atomicAdd(ap + 0, hv.x * nrm);
  atomicAdd(ap + 1, hv.y * nrm);
  atomicAdd(ap + 2, hv.z * nrm);
  atomicAdd(ap + 3, hv.w * nrm);
}

// ---------------------------------------------------------------------------
// Fused self-loop + ReLU: x = max(agg + h*dinv^2, 0)
// ---------------------------------------------------------------------------
__global__ void k_selfloop_relu(const float* __restrict__ H, const float* __restrict__ agg,
                                const float* __restrict__ dinv, float* __restrict__ X, long tot) {
  long i = (long)blockIdx.x * blockDim.x + threadIdx.x;
  if (i >= tot) return;
  const int node = (int)(i >> 6);
  const float di = dinv[node];
  const float v = agg[i] + H[i] * di * di;
  X[i] = v > 0.0f ? v : 0.0f;
}

// ---------------------------------------------------------------------------
extern "C" void kernel_launch(void* const* d_in, const int* in_sizes, int n_in,
                              void* d_out, int out_size, void* d_ws, size_t ws_size,
                              hipStream_t stream) {
  const float* x  = (const float*)d_in[0];
  const int*   ei = (const int*)d_in[1];          // [2, E] int32
  const float* ew = (const float*)d_in[2];
  const float* Wl[3] = {(const float*)d_in[3], (const float*)d_in[5], (const float*)d_in[7]};
  const float* bl[3] = {(const float*)d_in[4], (const float*)d_in[6], (const float*)d_in[8]};

  const int N = in_sizes[0] / DF;                 // 50000
  const int E = in_sizes[1] / 2;                  // 800000
  const long tot = (long)N * DF;

  // Workspace carve-up (16B-aligned slices): dinv[N] | H[N*64] | agg[N*64]
  float* dinv = (float*)d_ws;
  float* H    = dinv + (((long)N + 255) & ~255L);
  float* agg  = H + tot;
  float* xout = (float*)d_out;

  const int T = 256;

  // --- gcn_norm ---
  k_init_deg <<<(N + T - 1) / T, T, 0, stream>>>(dinv, N);
  k_accum_deg<<<(E + T - 1) / T, T, 0, stream>>>(ei, ew, dinv, E);
  k_dinv     <<<(N + T - 1) / T, T, 0, stream>>>(dinv, N);

  // --- 3 GCN layers ---
  const float* xin = x;
  const long et = (long)E * 16;                   // 16 threads per edge
  for (int l = 0; l < 3; ++l) {
    k_gemm_wmma    <<<(N + 127) / 128, 256, 0, stream>>>(xin, Wl[l], H, N);
    k_init_agg     <<<(int)((tot + T - 1) / T), T, 0, stream>>>(agg, bl[l], tot);
    k_scatter_edges<<<(int)((et + T - 1) / T), T, 0, stream>>>(ei, ew, dinv, H, agg, E);
    k_selfloop_relu<<<(int)((tot + T - 1) / T), T, 0, stream>>>(H, agg, dinv, xout, tot);
    xin = xout;                                   // ping through d_out
  }
}